// QuantumLayer_37014028157281
// MI455X (gfx1250) — compile-verified
//
#include <hip/hip_runtime.h>
#include <cmath>

// QuantumLayer closed form (derived analytically from the reference circuit):
//   e0 = 0   (q0 coherence destroyed by CZ against q3 in |+>; rho_0 = I/2)
//   e1 = -cos^2(1) * sin( (pi/2) * (1 + w1) )
//   e2 = 0   (only diagonal / population-preserving ops touch q2)
//   e3 = 0   (only diagonal CZ touches q3; populations stay 1/2,1/2)
// where w1 = inputs[b, 4].  w0 = inputs[b, 3] provably drops out (any unitary
// on the maximally-mixed q0 leaves all measured observables unchanged).
//
// Pure HBM-streaming map: 36 MB total traffic (~1.55 us at 23.3 TB/s).
// CDNA5 tuning: wave32 blocks, 4 rows per thread so all traffic is b128,
// non-temporal TH hints so the one-pass stream does not pollute L2.

typedef __attribute__((ext_vector_type(4))) float v4f;  // native vector: OK for NT builtins

__global__ __launch_bounds__(256) void qexp_quad_kernel(
    const v4f* __restrict__ in4,      // v4f view of (B,5) float32
    v4f* __restrict__ out4,           // v4f view of (B,4) float32
    float scale,                      // -cos^2(1)
    float halfpi,                     // pi/2
    int nquads)                       // B/4
{
    int t = blockIdx.x * blockDim.x + threadIdx.x;
    if (t >= nquads) return;

    // Rows 4t..4t+3 occupy floats [20t, 20t+20) = v4f slots [5t, 5t+5).
    // w1 of row 4t+r sits at flat index 20t + 5r + 4:
    //   r=0 -> slot 5t+1 comp 0   r=1 -> slot 5t+2 comp 1
    //   r=2 -> slot 5t+3 comp 2   r=3 -> slot 5t+4 comp 3
    const v4f v1 = __builtin_nontemporal_load(in4 + 5 * t + 1);
    const v4f v2 = __builtin_nontemporal_load(in4 + 5 * t + 2);
    const v4f v3 = __builtin_nontemporal_load(in4 + 5 * t + 3);
    const v4f v4 = __builtin_nontemporal_load(in4 + 5 * t + 4);

    // e1 = scale * sin(halfpi*(1+w1)); angle in [pi/2, pi) -> hw sin is accurate.
    const float e1_0 = scale * __sinf(fmaf(halfpi, v1[0], halfpi));
    const float e1_1 = scale * __sinf(fmaf(halfpi, v2[1], halfpi));
    const float e1_2 = scale * __sinf(fmaf(halfpi, v3[2], halfpi));
    const float e1_3 = scale * __sinf(fmaf(halfpi, v4[3], halfpi));

    const v4f o0 = {0.0f, e1_0, 0.0f, 0.0f};
    const v4f o1 = {0.0f, e1_1, 0.0f, 0.0f};
    const v4f o2 = {0.0f, e1_2, 0.0f, 0.0f};
    const v4f o3 = {0.0f, e1_3, 0.0f, 0.0f};

    __builtin_nontemporal_store(o0, out4 + 4 * t + 0);
    __builtin_nontemporal_store(o1, out4 + 4 * t + 1);
    __builtin_nontemporal_store(o2, out4 + 4 * t + 2);
    __builtin_nontemporal_store(o3, out4 + 4 * t + 3);
}

// Scalar tail for B % 4 != 0 (not hit for B = 1<<20, kept for robustness).
__global__ void qexp_tail_kernel(
    const float* __restrict__ in, float* __restrict__ out,
    float scale, float halfpi, int base, int n)
{
    int i = base + blockIdx.x * blockDim.x + threadIdx.x;
    if (i >= n) return;
    const float w1 = in[5 * i + 4];
    const float e1 = scale * __sinf(fmaf(halfpi, w1, halfpi));
    const v4f o = {0.0f, e1, 0.0f, 0.0f};
    __builtin_nontemporal_store(o, reinterpret_cast<v4f*>(out) + i);  // rows 16B-aligned
}

extern "C" void kernel_launch(void* const* d_in, const int* in_sizes, int n_in,
                              void* d_out, int out_size, void* d_ws, size_t ws_size,
                              hipStream_t stream)
{
    const float* in = (const float*)d_in[0];
    float* out = (float*)d_out;

    const int n = in_sizes[0] / 5;      // batch rows
    const int quads = n / 4;
    const int rem = n - 4 * quads;

    const double c1 = std::cos(1.0);
    const float scale = (float)(-(c1 * c1));            // -cos^2(1)
    const float halfpi = 1.57079632679489662f;          // pi/2

    if (quads > 0) {
        const int threads = 256;                        // 8 wave32 waves
        const int blocks = (quads + threads - 1) / threads;
        qexp_quad_kernel<<<blocks, threads, 0, stream>>>(
            (const v4f*)in, (v4f*)out, scale, halfpi, quads);
    }
    if (rem > 0) {
        qexp_tail_kernel<<<1, 32, 0, stream>>>(in, out, scale, halfpi, 4 * quads, n);
    }
}